// TIMBREGNN_24910810317306
// MI455X (gfx1250) — compile-verified
//
#include <hip/hip_runtime.h>
#include <math.h>

typedef __attribute__((ext_vector_type(2))) float v2f;
typedef __attribute__((ext_vector_type(8))) float v8f;

#define HDIM 64

// ---------------------------------------------------------------------------
// Scatter-accumulate: msg[dst] += h_src[src], cnt[dst] += 1   (segment mean)
// 16 lanes per edge, float4 gather, HW fp32 global atomics (no CAS loop).
// ---------------------------------------------------------------------------
__global__ __launch_bounds__(256)
void gnn_scatter_accum(const float* __restrict__ hsrc,
                       const int* __restrict__ src,
                       const int* __restrict__ dst,
                       float* __restrict__ msg,
                       float* __restrict__ cnt,
                       int nEdges) {
    const long long tid = (long long)blockIdx.x * blockDim.x + threadIdx.x;
    const long long e = tid >> 4;        // edge index
    const int q = (int)(tid & 15);       // 4-float chunk within 64-float row
    if (e >= nEdges) return;
    const int s = src[e];
    const int d = dst[e];
    const float4 v = *(const float4*)(hsrc + (long long)s * HDIM + q * 4);
    float* mrow = msg + (long long)d * HDIM + q * 4;
    unsafeAtomicAdd(mrow + 0, v.x);
    unsafeAtomicAdd(mrow + 1, v.y);
    unsafeAtomicAdd(mrow + 2, v.z);
    unsafeAtomicAdd(mrow + 3, v.w);
    if (q == 0) unsafeAtomicAdd(cnt + d, 1.0f);
}

// ---------------------------------------------------------------------------
// Fused SAGE tile kernel (one wave == one 16-row tile of dst nodes):
//   D = (msg/cnt) @ Wl^T + x_dst @ Wr^T + bl       (V_WMMA_F32_16X16X4_F32)
//   out = LayerNorm(gelu_erf(D)) * gamma + beta
// 8 waves / block. No cross-wave dependencies -> no block barriers.
// ---------------------------------------------------------------------------
__global__ __launch_bounds__(256)
void gnn_sage_wmma(const float* __restrict__ msg,
                   const float* __restrict__ cnt,
                   const float* __restrict__ xdst,
                   const float* __restrict__ Wl,    // [64][64], row = out-feature
                   const float* __restrict__ bl,    // [64]
                   const float* __restrict__ Wr,    // [64][64]
                   const float* __restrict__ gamma, // [64]
                   const float* __restrict__ beta,  // [64]
                   float* __restrict__ out,
                   int nDst) {
    __shared__ float lds[8][16][68];     // per-wave tile, padded vs bank conflicts

    const int wave = threadIdx.x >> 5;
    const int lane = threadIdx.x & 31;
    const int half = lane >> 4;          // 0: K-pair {0,1}/rows 0-7 | 1: {2,3}/rows 8-15
    const int ln   = lane & 15;

    const int tile = blockIdx.x * 8 + wave;
    const int row0 = tile * 16;
    if (row0 >= nDst) return;            // wave-local exit; no barriers below

    const int m = row0 + ln;             // this lane's A-matrix row
    const float ic = 1.0f / fmaxf(cnt[m], 1.0f);

    v8f acc[4];
    acc[0] = 0.0f; acc[1] = 0.0f; acc[2] = 0.0f; acc[3] = 0.0f;

    const float* arowL = msg  + (long long)m * HDIM + 2 * half;
    const float* arowR = xdst + (long long)m * HDIM + 2 * half;

    // pass 1: mean-aggregated neighbors x Wl^T
#pragma unroll
    for (int k0 = 0; k0 < HDIM; k0 += 4) {
        v2f a = *(const v2f*)(arowL + k0);
        a[0] *= ic; a[1] *= ic;
#pragma unroll
        for (int nt = 0; nt < 4; ++nt) {
            const int col = nt * 16 + ln;                       // output feature
            v2f b = *(const v2f*)(Wl + col * HDIM + k0 + 2 * half);
            acc[nt] = __builtin_amdgcn_wmma_f32_16x16x4_f32(
                false, a, false, b, (short)0, acc[nt], false, false);
        }
    }
    // pass 2: x_dst x Wr^T, accumulated into same C
#pragma unroll
    for (int k0 = 0; k0 < HDIM; k0 += 4) {
        v2f a = *(const v2f*)(arowR + k0);
#pragma unroll
        for (int nt = 0; nt < 4; ++nt) {
            const int col = nt * 16 + ln;
            v2f b = *(const v2f*)(Wr + col * HDIM + k0 + 2 * half);
            acc[nt] = __builtin_amdgcn_wmma_f32_16x16x4_f32(
                false, a, false, b, (short)0, acc[nt], false, false);
        }
    }

    // epilogue: bias + exact-erf GELU -> LDS tile
    // D layout: VGPR r holds M = r + 8*half (lanes 0-15 / 16-31), N = nt*16 + ln
#pragma unroll
    for (int nt = 0; nt < 4; ++nt) {
        const int n = nt * 16 + ln;
        const float bn = bl[n];
#pragma unroll
        for (int r = 0; r < 8; ++r) {
            const int mr = r + half * 8;
            const float v = acc[nt][r] + bn;
            const float g = 0.5f * v * (1.0f + erff(v * 0.70710678118654752f));
            lds[wave][mr][n] = g;
        }
    }
    asm volatile("s_wait_dscnt 0" ::: "memory");   // wave-local LDS RAW fence

    // LayerNorm: lanes (ln, ln+16) cooperate on row ln, 32 columns each
    const int rrow = ln;
    float s = 0.0f, sq = 0.0f;
#pragma unroll
    for (int j = 0; j < 32; ++j) {
        const float v = lds[wave][rrow][half * 32 + j];
        s += v; sq += v * v;
    }
    s  += __shfl_xor(s, 16);
    sq += __shfl_xor(sq, 16);
    const float mean = s * (1.0f / 64.0f);
    const float var  = sq * (1.0f / 64.0f) - mean * mean;
    const float rstd = rsqrtf(var + 1e-5f);

    float* orow = out + (long long)(row0 + rrow) * HDIM + half * 32;
#pragma unroll
    for (int j = 0; j < 32; j += 4) {
        const int n = half * 32 + j;
        float4 o;
        o.x = (lds[wave][rrow][n + 0] - mean) * rstd * gamma[n + 0] + beta[n + 0];
        o.y = (lds[wave][rrow][n + 1] - mean) * rstd * gamma[n + 1] + beta[n + 1];
        o.z = (lds[wave][rrow][n + 2] - mean) * rstd * gamma[n + 2] + beta[n + 2];
        o.w = (lds[wave][rrow][n + 3] - mean) * rstd * gamma[n + 3] + beta[n + 3];
        *(float4*)(orow + j) = o;
    }
}

// ---------------------------------------------------------------------------
extern "C" void kernel_launch(void* const* d_in, const int* in_sizes, int n_in,
                              void* d_out, int out_size, void* d_ws, size_t ws_size,
                              hipStream_t stream) {
    const float* emb_sl  = (const float*)d_in[0];
    const float* emb_api = (const float*)d_in[1];
    const float* Wl      = (const float*)d_in[2];   // [L,2,64,64]
    const float* bl      = (const float*)d_in[3];   // [L,2,64]
    const float* Wr      = (const float*)d_in[4];   // [L,2,64,64]
    const float* gamma   = (const float*)d_in[5];   // [L,64]
    const float* beta    = (const float*)d_in[6];   // [L,64]
    const int* src_sa    = (const int*)d_in[7];
    const int* dst_sa    = (const int*)d_in[8];
    const int* src_as    = (const int*)d_in[9];
    const int* dst_as    = (const int*)d_in[10];

    const int N_SL  = in_sizes[0] / HDIM;
    const int N_API = in_sizes[1] / HDIM;
    const int L     = in_sizes[5] / HDIM;
    const int E     = in_sizes[7];

    // workspace carve-up (floats)
    float* ws       = (float*)d_ws;
    float* msg_sl   = ws;                                   // N_SL*64
    float* msg_api  = msg_sl  + (size_t)N_SL  * HDIM;       // N_API*64
    float* cnt_sl   = msg_api + (size_t)N_API * HDIM;       // N_SL
    float* cnt_api  = cnt_sl  + (size_t)N_SL;               // N_API
    float* h_sl_t   = cnt_api + (size_t)N_API;              // N_SL*64 (ping-pong)
    float* h_api_t  = h_sl_t  + (size_t)N_SL  * HDIM;       // N_API*64

    float* dout_sl  = (float*)d_out;                        // [N_SL,64]
    float* dout_api = dout_sl + (size_t)N_SL * HDIM;        // [N_API,64]

    // edge-phase launch geometry: 16 lanes per edge
    const long long scatter_threads = (long long)E * 16;
    const int scatter_blocks = (int)((scatter_threads + 255) / 256);

    const int tiles_api  = N_API / 16;
    const int blocks_api = (tiles_api + 7) / 8;
    const int tiles_sl   = N_SL / 16;
    const int blocks_sl  = (tiles_sl + 7) / 8;

    const float* cur_sl  = emb_sl;
    const float* cur_api = emb_api;

    for (int l = 0; l < L; ++l) {
        // buffer schedule: L0 -> d_out, L1 -> ws tmp, L2 -> d_out (final)
        float* nx_sl  = (l == 1) ? h_sl_t  : dout_sl;
        float* nx_api = (l == 1) ? h_api_t : dout_api;

        // ---- edge type 0: shortlist -> api ----
        hipMemsetAsync(msg_api, 0, (size_t)N_API * HDIM * sizeof(float), stream);
        hipMemsetAsync(cnt_api, 0, (size_t)N_API * sizeof(float), stream);
        gnn_scatter_accum<<<scatter_blocks, 256, 0, stream>>>(
            cur_sl, src_sa, dst_sa, msg_api, cnt_api, E);
        gnn_sage_wmma<<<blocks_api, 256, 0, stream>>>(
            msg_api, cnt_api, cur_api,
            Wl + (size_t)(l * 2 + 0) * HDIM * HDIM,
            bl + (size_t)(l * 2 + 0) * HDIM,
            Wr + (size_t)(l * 2 + 0) * HDIM * HDIM,
            gamma + (size_t)l * HDIM, beta + (size_t)l * HDIM,
            nx_api, N_API);

        // ---- edge type 1: api -> shortlist (reads untouched cur buffers) ----
        hipMemsetAsync(msg_sl, 0, (size_t)N_SL * HDIM * sizeof(float), stream);
        hipMemsetAsync(cnt_sl, 0, (size_t)N_SL * sizeof(float), stream);
        gnn_scatter_accum<<<scatter_blocks, 256, 0, stream>>>(
            cur_api, src_as, dst_as, msg_sl, cnt_sl, E);
        gnn_sage_wmma<<<blocks_sl, 256, 0, stream>>>(
            msg_sl, cnt_sl, cur_sl,
            Wl + (size_t)(l * 2 + 1) * HDIM * HDIM,
            bl + (size_t)(l * 2 + 1) * HDIM,
            Wr + (size_t)(l * 2 + 1) * HDIM * HDIM,
            gamma + (size_t)l * HDIM, beta + (size_t)l * HDIM,
            nx_sl, N_SL);

        cur_sl  = nx_sl;
        cur_api = nx_api;
    }
}